// InputProcess_wGCN_87205015978499
// MI455X (gfx1250) — compile-verified
//
#include <hip/hip_runtime.h>

typedef float v2f __attribute__((ext_vector_type(2)));
typedef float v8f __attribute__((ext_vector_type(8)));

#define NT    196   // frames
#define DIN   263   // joints*feats
#define DOUT  256   // latent
#define BATCH 1024
#define SUPS  264   // LDS row stride (256 + 8 pad: 2*264 % 64banks = 16 -> halves disjoint)
#define MT    13    // ceil(196/16) M-tiles
#define NGRP  2     // 2 N-groups of 8 tiles (8*16 = 128 cols each)
#define NUNITS (MT * NGRP)  // 26
#define KMAIN 260   // 65 full k-steps of 4; tail handles j = 260..262 (+ zero for 263)

__global__ __launch_bounds__(256)
void gcn_fused_kernel(const float* __restrict__ x,     // [1024, 263, 1, 196]
                      const float* __restrict__ W,     // [263, 256]
                      const float* __restrict__ att,   // [196, 196]
                      const float* __restrict__ bias,  // [256]
                      float* __restrict__ out)         // [196, 1024, 256]
{
    __shared__ float sup[NT * SUPS];   // 206,976 B of 320 KB WGP LDS

    const int b    = blockIdx.x;
    const int lane = threadIdx.x & 31;
    const int wave = threadIdx.x >> 5;   // 0..7
    const int half = lane >> 4;          // 0: lanes 0-15, 1: lanes 16-31
    const int l16  = lane & 15;
    const int kh   = 2 * half;           // K offset held by this lane-half (v0; v1 = +1)

    const float* __restrict__ xb = x + (size_t)b * DIN * NT;

    // ---------------- Phase 1: sup = xr[b] @ W ----------------
    // xr[b][t][j] == xb[j*196 + t]  (t contiguous -> coalesced A loads)
    for (int u = wave; u < NUNITS; u += 8) {
        const int m0 = (u % MT) * 16;
        const int nb = (u / MT) * 128;   // N-group column base

        v8f acc[8];
        #pragma unroll
        for (int i = 0; i < 8; ++i)
            #pragma unroll
            for (int r = 0; r < 8; ++r) acc[i][r] = 0.0f;

        const int t  = m0 + l16;
        const int tc = (t < NT) ? t : (NT - 1);  // clamp only; garbage rows never stored
        const float* __restrict__ aCol  = xb + tc;        // + j*NT per element
        const float* __restrict__ wBase = W + nb + l16;   // + j*DOUT + i*16

        // Main loop: all j in [0, 259] -> branch-free
        for (int k0 = 0; k0 < KMAIN; k0 += 4) {
            const int j0 = k0 + kh;
            v2f a;
            a.x = aCol[j0 * NT];
            a.y = aCol[(j0 + 1) * NT];

            v2f bf[8];
            #pragma unroll
            for (int i = 0; i < 8; ++i) {
                bf[i].x = wBase[j0 * DOUT + i * 16];
                bf[i].y = wBase[(j0 + 1) * DOUT + i * 16];
            }
            #pragma unroll
            for (int i = 0; i < 8; ++i)
                acc[i] = __builtin_amdgcn_wmma_f32_16x16x4_f32(
                    false, a, false, bf[i], (short)0, acc[i], false, false);
        }

        // Tail step: j0 = 260|262, j1 = 261|263 (263 is out of range -> zero A and B)
        {
            const int j0 = KMAIN + kh;
            const int j1 = j0 + 1;
            const bool v1ok = (j1 < DIN);

            v2f a;
            a.x = aCol[j0 * NT];
            a.y = v1ok ? aCol[j1 * NT] : 0.0f;

            v2f bf[8];
            #pragma unroll
            for (int i = 0; i < 8; ++i) {
                bf[i].x = wBase[j0 * DOUT + i * 16];
                bf[i].y = v1ok ? wBase[j1 * DOUT + i * 16] : 0.0f;
            }
            #pragma unroll
            for (int i = 0; i < 8; ++i)
                acc[i] = __builtin_amdgcn_wmma_f32_16x16x4_f32(
                    false, a, false, bf[i], (short)0, acc[i], false, false);
        }

        // D layout: VGPR r -> rows (m0 + r) [lanes 0-15] and (m0 + r + 8) [lanes 16-31]
        #pragma unroll
        for (int i = 0; i < 8; ++i) {
            const int n = nb + i * 16 + l16;
            #pragma unroll
            for (int r = 0; r < 8; ++r) {
                const int row = m0 + r + 8 * half;
                if (row < NT) sup[row * SUPS + n] = acc[i][r];
            }
        }
    }

    __syncthreads();

    // ---------------- Phase 2: out = att @ sup + bias ----------------
    // K = 196 = 49 * 4 exactly -> fully branch-free K-loop.
    for (int u = wave; u < NUNITS; u += 8) {
        const int m0 = (u % MT) * 16;
        const int nb = (u / MT) * 128;

        const int t  = m0 + l16;
        const int tc = (t < NT) ? t : (NT - 1);
        const float* __restrict__ aRow = att + tc * NT + kh;

        v8f acc[8];
        #pragma unroll
        for (int i = 0; i < 8; ++i) {
            const float bv = bias[nb + i * 16 + l16];  // column-only dependence
            #pragma unroll
            for (int r = 0; r < 8; ++r) acc[i][r] = bv;
        }

        for (int k0 = 0; k0 < NT; k0 += 4) {
            const int s0 = k0 + kh;

            v2f a;
            a.x = aRow[k0];
            a.y = aRow[k0 + 1];

            v2f bf[8];
            #pragma unroll
            for (int i = 0; i < 8; ++i) {
                const int n = nb + i * 16 + l16;
                bf[i].x = sup[s0 * SUPS + n];
                bf[i].y = sup[(s0 + 1) * SUPS + n];
            }
            #pragma unroll
            for (int i = 0; i < 8; ++i)
                acc[i] = __builtin_amdgcn_wmma_f32_16x16x4_f32(
                    false, a, false, bf[i], (short)0, acc[i], false, false);
        }

        // out[t, b, d] with d contiguous -> coalesced half-wave 64B stores
        #pragma unroll
        for (int i = 0; i < 8; ++i) {
            const int n = nb + i * 16 + l16;
            #pragma unroll
            for (int r = 0; r < 8; ++r) {
                const int row = m0 + r + 8 * half;
                if (row < NT)
                    out[((size_t)row * BATCH + b) * DOUT + n] = acc[i][r];
            }
        }
    }
}

extern "C" void kernel_launch(void* const* d_in, const int* in_sizes, int n_in,
                              void* d_out, int out_size, void* d_ws, size_t ws_size,
                              hipStream_t stream) {
    const float* x    = (const float*)d_in[0];  // [1024, 263, 1, 196]
    const float* W    = (const float*)d_in[1];  // [263, 256]
    const float* att  = (const float*)d_in[2];  // [196, 196]
    const float* bias = (const float*)d_in[3];  // [256]
    float* out        = (float*)d_out;          // [196, 1024, 256]

    gcn_fused_kernel<<<BATCH, 256, 0, stream>>>(x, W, att, bias, out);
}